// FASTMultiHeadAttention_40458591928588
// MI455X (gfx1250) — compile-verified
//
#include <hip/hip_runtime.h>

typedef __attribute__((ext_vector_type(2))) float v2f;
typedef __attribute__((ext_vector_type(8))) float v8f;

// D = A(16x4) * B(4x16) + C, fp32 throughout.
#define WMMA_F32(c, a, b) \
  (c) = __builtin_amdgcn_wmma_f32_16x16x4_f32(false, (a), false, (b), (short)0, (c), false, false)

constexpr int BH     = 16;        // b*h
constexpr int N      = 8192;
constexpr int D      = 32;
constexpr int SEG    = 16;        // segments per head (16*16 = 256 WGs)
constexpr int SEGLEN = N / SEG;   // 512
constexpr int CH     = 16;        // chunk rows (one M tile)
constexpr int NCH    = SEGLEN / CH;
constexpr int DP     = 48;        // padded feature dim: 32 data | 1 ones | 15 zeros
constexpr int LS     = 52;        // row stride, normal tiles (mult of 4 dw: B128-aligned; conflict-free)
constexpr int TS     = 18;        // row stride, transposed tiles (even: B64 reads aligned)
constexpr int SDIM   = 33;        // compact state dim in workspace

__device__ inline unsigned lds_off(const void* p) {   // generic->LDS byte offset
  return (unsigned)(uintptr_t)p;
}
__device__ inline void async_b128(unsigned lds, const float* g) {
  asm volatile("global_load_async_to_lds_b128 %0, %1, off" :: "v"(lds), "v"(g) : "memory");
}
__device__ inline void async_b32(unsigned lds, const float* g) {
  asm volatile("global_load_async_to_lds_b32 %0, %1, off" :: "v"(lds), "v"(g) : "memory");
}
__device__ inline void wait_async0() {
  asm volatile("s_wait_asynccnt 0x0" ::: "memory");
}

// ---------------- Pass 1: per-segment local state  S_seg = sum k_hat v_hat^T ----
__global__ __launch_bounds__(96)
void fastmax_pass1(const float* __restrict__ Kg, const float* __restrict__ Vg,
                   float* __restrict__ Sloc) {
  __shared__ float shKT[2][DP * TS];   // K^T: [dim][row], double buffered
  __shared__ float shVT[2][DP * TS];   // V^T: [dim][row], double buffered
  const int seg = blockIdx.x, bh = blockIdx.y;
  const int tid = threadIdx.x;
  const int wave = tid >> 5, lane = tid & 31;
  const int half = lane >> 4, ln = lane & 15;
  const long base = (long)bh * N + (long)seg * SEGLEN;

  // augmentation rows (dims 32..47): dim32 = 1, rest 0 — written once, both buffers
  for (int i = tid; i < (DP - D) * CH; i += 96) {
    int dm = D + i / CH, r = i % CH;
    float pv = (dm == D) ? 1.0f : 0.0f;
    shKT[0][dm * TS + r] = pv; shKT[1][dm * TS + r] = pv;
    shVT[0][dm * TS + r] = pv; shVT[1][dm * TS + r] = pv;
  }

  // async element-scatter copy of one chunk (transposing), straight-line per matrix
  auto issue = [&](int ch, int buf) {
    const float* gk = Kg + (base + (long)ch * CH) * D;
    const float* gv = Vg + (base + (long)ch * CH) * D;
    for (int i = tid; i < CH * D; i += 96) {
      int r = i >> 5, c = i & 31;
      async_b32(lds_off(&shKT[buf][c * TS + r]), gk + r * D + c);
    }
    for (int i = tid; i < CH * D; i += 96) {
      int r = i >> 5, c = i & 31;
      async_b32(lds_off(&shVT[buf][c * TS + r]), gv + r * D + c);
    }
  };

  v8f sacc[3];
  for (int mt = 0; mt < 3; ++mt) sacc[mt] = {};

  issue(0, 0);
  for (int ch = 0; ch < NCH; ++ch) {
    const int cur = ch & 1;
    wait_async0();
    __syncthreads();
    if (ch + 1 < NCH) issue(ch + 1, cur ^ 1);
    const v2f* kt2 = (const v2f*)shKT[cur];
    const v2f* vt2 = (const v2f*)shVT[cur];
    // S(48x16-tile 'wave') += Khat^T(48x16) * Vhat(16x48-tile), K=16 -> 4 steps
    for (int kk = 0; kk < 4; ++kk) {
      v2f b = vt2[(16 * wave + ln) * (TS / 2) + 2 * kk + half];
      for (int mt = 0; mt < 3; ++mt) {
        v2f a = kt2[(16 * mt + ln) * (TS / 2) + 2 * kk + half];
        WMMA_F32(sacc[mt], a, b);
      }
    }
    __syncthreads();   // compute done before next buffer reuse
  }
  // store compact 33x33 state (rows/cols >=33 are identically zero)
  int n = 16 * wave + ln;
  if (n < SDIM) {
    for (int mt = 0; mt < 3; ++mt)
      for (int r = 0; r < 8; ++r) {
        int m = 16 * mt + r + 8 * half;            // C/D layout: m = vgpr + 8*half
        if (m < SDIM)
          Sloc[(((long)bh * SEG + seg) * SDIM + m) * SDIM + n] = sacc[mt][r];
      }
  }
}

// ---------------- Pass 2: in-place exclusive scan of segment states ------------
__global__ __launch_bounds__(128)
void fastmax_pass2(float* __restrict__ Sbuf) {
  const int bh = blockIdx.x;
  for (int e = threadIdx.x; e < SDIM * SDIM; e += 128) {
    float run = 0.0f;
    for (int s = 0; s < SEG; ++s) {
      long idx = ((long)bh * SEG + s) * (SDIM * SDIM) + e;
      float t = Sbuf[idx];
      Sbuf[idx] = run;
      run += t;
    }
  }
}

// ---------------- Pass 3: chunked causal linear attention ----------------------
__global__ __launch_bounds__(96)
void fastmax_pass3(const float* __restrict__ Qg, const float* __restrict__ Kg,
                   const float* __restrict__ Vg, const float* __restrict__ Spre,
                   float* __restrict__ Out) {
  __shared__ float shQ[2][CH * LS];     // Qhat, row-major
  __shared__ float shK[2][CH * LS];     // Khat, row-major (B operand of Q*K^T)
  __shared__ float shKT[2][DP * TS];    // Khat^T (A operand of state update)
  __shared__ float shVT[2][DP * TS];    // Vhat^T (B operand pairs contiguous)
  __shared__ float lden[CH];
  __shared__ float lA[3][CH * TS];      // per-wave masked-A scratch (D->A relayout)
  __shared__ float lS[3][CH * LS];      // per-wave state mirror, [n][m] (D->B relayout)

  const int seg = blockIdx.x, bh = blockIdx.y;
  const int tid = threadIdx.x;
  const int wave = tid >> 5, lane = tid & 31;
  const int half = lane >> 4, ln = lane & 15;
  const long base = (long)bh * N + (long)seg * SEGLEN;
  float* As = lA[wave];
  float* Sm = lS[wave];
  const v2f* as2 = (const v2f*)As;
  const v2f* sm2 = (const v2f*)Sm;

  // init running state from exclusive prefix (D-format regs + LDS mirror [n][m])
  v8f sacc[3];
  {
    int n = 16 * wave + ln;
    for (int mt = 0; mt < 3; ++mt) {
      v8f c = {};
      for (int r = 0; r < 8; ++r) {
        int m = 16 * mt + r + 8 * half;
        float v = 0.0f;
        if (m < SDIM && n < SDIM)
          v = Spre[(((long)bh * SEG + seg) * SDIM + m) * SDIM + n];
        c[r] = v;
      }
      sacc[mt] = c;
      for (int r = 0; r < 8; ++r)
        Sm[ln * LS + (16 * mt + r + 8 * half)] = c[r];
    }
  }
  // augmentation padding, written once for both buffers
  for (int i = tid; i < CH * (DP - D); i += 96) {
    int r = i / (DP - D), c = D + i % (DP - D);
    float pv = (c == D) ? 1.0f : 0.0f;
    shQ[0][r * LS + c] = pv;  shQ[1][r * LS + c] = pv;
    shK[0][r * LS + c] = pv;  shK[1][r * LS + c] = pv;
    shKT[0][c * TS + r] = pv; shKT[1][c * TS + r] = pv;
    shVT[0][c * TS + r] = pv; shVT[1][c * TS + r] = pv;
  }

  // async copy of one chunk: straight-line per-matrix loops (no pointer selects)
  auto issue = [&](int ch, int buf) {
    const float* gq = Qg + (base + (long)ch * CH) * D;
    const float* gk = Kg + (base + (long)ch * CH) * D;
    const float* gv = Vg + (base + (long)ch * CH) * D;
    for (int i = tid; i < CH * (D / 4); i += 96) {       // Q row-major, 4 floats/lane
      int r = i >> 3, c4 = (i & 7) * 4;
      async_b128(lds_off(&shQ[buf][r * LS + c4]), gq + r * D + c4);
    }
    for (int i = tid; i < CH * (D / 4); i += 96) {       // K row-major
      int r = i >> 3, c4 = (i & 7) * 4;
      async_b128(lds_off(&shK[buf][r * LS + c4]), gk + r * D + c4);
    }
    for (int i = tid; i < CH * D; i += 96) {             // K^T element scatter
      int r = i >> 5, c = i & 31;
      async_b32(lds_off(&shKT[buf][c * TS + r]), gk + r * D + c);
    }
    for (int i = tid; i < CH * D; i += 96) {             // V^T element scatter
      int r = i >> 5, c = i & 31;
      async_b32(lds_off(&shVT[buf][c * TS + r]), gv + r * D + c);
    }
  };

  issue(0, 0);
  for (int ch = 0; ch < NCH; ++ch) {
    const int cur = ch & 1;
    wait_async0();
    __syncthreads();
    if (ch + 1 < NCH) issue(ch + 1, cur ^ 1);
    const v2f* q2  = (const v2f*)shQ[cur];
    const v2f* k2  = (const v2f*)shK[cur];
    const v2f* kt2 = (const v2f*)shKT[cur];
    const v2f* vt2 = (const v2f*)shVT[cur];

    // Fused K=48 sweep, shared Qhat A-fragments:
    //   A  = Qhat * Khat^T            (16x16, +J via augmented ones column)
    //   Cn = Qhat * S_prev            (numer inter-chunk part)
    v8f A = {}, Cn = {};
    for (int kk = 0; kk < 12; ++kk) {
      int idx = ln * (LS / 2) + 2 * kk + half;
      v2f a  = q2[idx];
      v2f bk = k2[idx];                 // B(k,n) = Khat[n][k], contiguous pair
      v2f bs = sm2[idx];                // B(k,n) = S_prev[k][n], mirror is [n][k]
      WMMA_F32(A, a, bk);
      WMMA_F32(Cn, a, bs);
    }
    // causal mask (element (m = r+8*half, n = ln)); stash in A-operand scratch
    {
      int m0 = 8 * half;
      for (int r = 0; r < 8; ++r)
        if (ln > m0 + r) A[r] = 0.0f;
      for (int r = 0; r < 8; ++r)
        As[(m0 + r) * TS + ln] = A[r];
    }

    // Fused K=16 sweep, shared Vhat B-fragments:
    //   Cn   += tril(A) * Vhat        (numer intra-chunk part; col 32 = denom)
    //   sacc += Khat^T * Vhat         (state update)
    for (int kk = 0; kk < 4; ++kk) {
      v2f b  = vt2[(16 * wave + ln) * (TS / 2) + 2 * kk + half];
      v2f aA = as2[ln * (TS / 2) + 2 * kk + half];
      WMMA_F32(Cn, aA, b);
      for (int mt = 0; mt < 3; ++mt) {
        v2f aK = kt2[(16 * mt + ln) * (TS / 2) + 2 * kk + half];
        WMMA_F32(sacc[mt], aK, b);
      }
    }

    // wave 2 owns cols 32..47: col 32 (local n==0) is the denominator
    if (wave == 2 && ln == 0)
      for (int r = 0; r < 8; ++r) lden[8 * half + r] = Cn[r];
    __syncthreads();
    if (wave < 2) {
      for (int r = 0; r < 8; ++r) {
        int m = 8 * half + r;
        long g = (base + (long)ch * CH + m) * D + 16 * wave + ln;
        Out[g] = Cn[r] / lden[m];
      }
    }

    // refresh per-wave state mirror ([n][m] so next chunk's pairs are contiguous)
    for (int mt = 0; mt < 3; ++mt)
      for (int r = 0; r < 8; ++r)
        Sm[ln * LS + (16 * mt + r + 8 * half)] = sacc[mt][r];
  }
}

extern "C" void kernel_launch(void* const* d_in, const int* in_sizes, int n_in,
                              void* d_out, int out_size, void* d_ws, size_t ws_size,
                              hipStream_t stream) {
  const float* q = (const float*)d_in[0];
  const float* k = (const float*)d_in[1];
  const float* v = (const float*)d_in[2];
  float* out  = (float*)d_out;
  float* sbuf = (float*)d_ws;   // BH*SEG*33*33 floats ~= 1.1 MB of scratch

  dim3 grid(SEG, BH);
  fastmax_pass1<<<grid, 96, 0, stream>>>(k, v, sbuf);
  fastmax_pass2<<<dim3(BH), 128, 0, stream>>>(sbuf);
  fastmax_pass3<<<grid, 96, 0, stream>>>(q, k, v, sbuf, out);
}